// LSResNet_59313498358232
// MI455X (gfx1250) — compile-verified
//
#include <hip/hip_runtime.h>
#include <math.h>

// Problem constants (from the reference)
constexpr int Bb = 8, Ss = 128, Vv = 200, Ww = 5, Kk = 80, NROT = 16;
constexpr int SPW  = 16;          // samples per workgroup (= WMMA M)
constexpr int BDIM = 800;         // 25 waves of 32: exactly one wave per (w, n_tile) GEMM task
constexpr float EPSc = 1e-5f;
constexpr float TWO_PI_F = 6.28318530717958647692f;

typedef __attribute__((ext_vector_type(2))) float v2f;
typedef __attribute__((ext_vector_type(8))) float v8f;

__device__ __forceinline__ void atomicMaxF(float* addr, float val) {
    unsigned int* ua = (unsigned int*)addr;
    unsigned int old = *ua;
    while (__uint_as_float(old) < val) {
        unsigned int assumed = old;
        old = atomicCAS(ua, assumed, __float_as_uint(val));
        if (old == assumed) break;
    }
}

__global__ void init_out_kernel(float* __restrict__ out, int n) {
    int i = blockIdx.x * blockDim.x + threadIdx.x;
    if (i < n) out[i] = -INFINITY;
}

__global__ __launch_bounds__(BDIM)
void moconv_kernel(const float* __restrict__ feat,      // [B,S,V,W]
                   const float* __restrict__ rho,       // [B,S,V]
                   const float* __restrict__ theta,     // [B,S,V]
                   const float* __restrict__ mask,      // [B,S,V]
                   const float* __restrict__ mu_rho,    // [W,K]
                   const float* __restrict__ sig_rho,   // [W,K]
                   const float* __restrict__ mu_th,     // [W,K]
                   const float* __restrict__ sig_th,    // [W,K]
                   const float* __restrict__ Wc,        // [W,K,K]
                   const float* __restrict__ bc,        // [W,K]
                   float* __restrict__ out)             // [B,S,W,K]
{
    // LDS staging: ~134 KB, fits the 320 KB/WGP CDNA5 LDS.
    __shared__ float s_rho[SPW * Vv];
    __shared__ float s_th [SPW * Vv];
    __shared__ float s_m  [SPW * Vv];
    __shared__ float s_f  [SPW * Vv * Ww];
    __shared__ float s_mur[Ww * Kk], s_isr[Ww * Kk];
    __shared__ float s_mut[Ww * Kk], s_ist[Ww * Kk];
    __shared__ __align__(16) float s_desc[SPW * Ww * Kk];  // [s][w][k]

    const int tid = threadIdx.x;
    const int bs0 = blockIdx.x * SPW;       // first flattened (b*S+s) sample
    const int rot = blockIdx.y;             // this workgroup's rotation

    // ---- stage inputs ----
    for (int i = tid; i < SPW * Vv; i += BDIM) {
        int gi = bs0 * Vv + i;
        s_rho[i] = rho[gi];
        s_th [i] = theta[gi];
        s_m  [i] = mask[gi];
    }
    for (int i = tid; i < SPW * Vv * Ww; i += BDIM)
        s_f[i] = feat[bs0 * Vv * Ww + i];
    for (int i = tid; i < Ww * Kk; i += BDIM) {
        s_mur[i] = mu_rho[i];
        float sr = sig_rho[i];
        s_isr[i] = 1.0f / (sr * sr + EPSc);
        s_mut[i] = mu_th[i];
        float st = sig_th[i];
        s_ist[i] = 1.0f / (st * st + EPSc);
    }
    __syncthreads();

    const float dth = (float)rot * (TWO_PI_F / (float)NROT);

    // ---- gaussian + vertex reduction: desc[s][w][k] ----
    // 6400 outputs / 800 threads = 8 per thread; 200-long v loop each.
    // Single fused exp: exp(a_rho)*exp(a_th) == exp(a_rho + a_th).
    for (int i = tid; i < SPW * Ww * Kk; i += BDIM) {
        const int s  = i / (Ww * Kk);
        const int wk = i - s * (Ww * Kk);
        const int w  = wk / Kk;
        const float mur = s_mur[wk], isr = s_isr[wk];
        const float mut = s_mut[wk], ist = s_ist[wk];
        const float* pr = &s_rho[s * Vv];
        const float* pt = &s_th [s * Vv];
        const float* pm = &s_m  [s * Vv];
        const float* pf = &s_f  [s * Vv * Ww + w];
        float sg = 0.0f, sgf = 0.0f;
        #pragma unroll 4
        for (int v = 0; v < Vv; ++v) {
            float dr = pr[v] - mur;
            float th = pt[v] + dth;                 // theta in [0,2pi), dth < 2pi
            th = (th >= TWO_PI_F) ? th - TWO_PI_F : th;   // == fmod(.., 2pi)
            float dt = th - mut;
            float g  = __expf(-(dr * dr * isr + dt * dt * ist)) * pm[v];
            sg  += g;
            sgf  = fmaf(g, pf[v * Ww], sgf);
        }
        s_desc[i] = sgf / (sg + EPSc);   // normalized weighted reduce
    }
    __syncthreads();

    // ---- GEMM via V_WMMA_F32_16X16X4_F32: 25 waves <-> (w, n_tile) ----
    const int wave = tid >> 5;
    const int lane = tid & 31;
    const int w    = wave / 5;
    const int nt   = wave % 5;
    const int oCol = nt * 16 + (lane & 15);     // N index (output channel)
    const int kHalf = (lane >> 4) << 1;         // A/B: lanes 16-31 carry K+2

    v8f c = {};
    #pragma unroll
    for (int kk = 0; kk < 20; ++kk) {           // K = 80 = 20 x 4
        const int kb = kk * 4 + kHalf;          // even -> 8B aligned LDS pair
        // A 16x4: lane%16 = M row (sample), VGPR pair = (kb, kb+1)
        v2f a = *(const v2f*)&s_desc[(lane & 15) * (Ww * Kk) + w * Kk + kb];
        // B 4x16: lane%16 = N col, VGPR pair = rows (kb, kb+1)
        v2f b;
        b.x = Wc[(w * Kk + kb    ) * Kk + oCol];
        b.y = Wc[(w * Kk + kb + 1) * Kk + oCol];
        c = __builtin_amdgcn_wmma_f32_16x16x4_f32(
                false, a, false, b, (short)0, c, false, false);
    }

    // bias + max-over-rotations (atomic fp32 max across the 16 rotation WGs)
    const float bias = bc[w * Kk + oCol];
    #pragma unroll
    for (int j = 0; j < 8; ++j) {
        // C/D layout: VGPR j -> M=j (lanes 0-15) / M=j+8 (lanes 16-31)
        const int sLoc = j + ((lane >> 4) << 3);
        const float val = c[j] + bias;
        atomicMaxF(&out[((bs0 + sLoc) * Ww + w) * Kk + oCol], val);
    }
}

extern "C" void kernel_launch(void* const* d_in, const int* in_sizes, int n_in,
                              void* d_out, int out_size, void* d_ws, size_t ws_size,
                              hipStream_t stream) {
    const float* feat    = (const float*)d_in[0];
    const float* rho     = (const float*)d_in[1];
    const float* theta   = (const float*)d_in[2];
    const float* mask    = (const float*)d_in[3];
    const float* mu_rho  = (const float*)d_in[4];
    const float* sig_rho = (const float*)d_in[5];
    const float* mu_th   = (const float*)d_in[6];
    const float* sig_th  = (const float*)d_in[7];
    const float* Wc      = (const float*)d_in[8];
    const float* bc      = (const float*)d_in[9];
    float* out = (float*)d_out;

    init_out_kernel<<<(out_size + 255) / 256, 256, 0, stream>>>(out, out_size);

    dim3 grid((Bb * Ss) / SPW, NROT);   // (64, 16)
    moconv_kernel<<<grid, BDIM, 0, stream>>>(feat, rho, theta, mask,
                                             mu_rho, sig_rho, mu_th, sig_th,
                                             Wc, bc, out);
}